// TFLiteSTFT_31473520345491
// MI455X (gfx1250) — compile-verified
//
#include <hip/hip_runtime.h>

typedef __attribute__((ext_vector_type(2))) float v2f;
typedef __attribute__((ext_vector_type(8))) float v8f;

#define FRAME_LEN   800
#define FRAME_STEP  200
#define SIG_LEN     240000
#define NUM_FRAMES  1197          // (240000-800)/200 + 1
#define NUM_BINS    401           // 800/2 + 1
#define KC          80            // K chunk staged in LDS
#define NCHUNK      10            // 800 / 80
#define LDS_STRIDE  84            // 80 + 4 pad floats -> conflict-free ds_load_b64
#define M_TILE      32            // frames per block (2 WMMA M-subtiles per wave)

// Block = 128 threads = 4 wave32. Tile: 32 frames (M) x 64 bins (N).
// Each wave owns a 32x16 bin sub-tile: two A fragments share each B fragment,
// so the cos/sin B loads are amortized 2x (1 global_load_b32 per v_wmma).
// Both real (cos) and imag (sin) accumulators are kept per M-subtile.
__global__ __launch_bounds__(128)
void stft_wmma_f32_kernel(const float* __restrict__ sig,
                          const float* __restrict__ win,
                          const float* __restrict__ dcos,
                          const float* __restrict__ dsin,
                          float* __restrict__ out)
{
    __shared__ float ldsA[M_TILE * LDS_STRIDE];

    const int tid  = threadIdx.x;
    const int lane = tid & 31;
    const int wave = tid >> 5;

    const int bin_tile = blockIdx.x;   // 0..6   (64 bins each, 448 >= 401)
    const int f_tile   = blockIdx.y;   // 0..37  (32 frames each, 1216 >= 1197)
    const int batch    = blockIdx.z;

    const int f_base = f_tile * M_TILE;
    const int bin    = bin_tile * 64 + wave * 16 + (lane & 15);

    const float* sigb = sig + (size_t)batch * SIG_LEN;

    // WMMA f32 16x16x4 fragment geometry (wave32):
    //  A (16x4): lanes 0-15 hold M=lane, K={0,1}; lanes 16-31 hold M=lane-16, K={2,3}
    //  B (4x16): VGPR0 = rows K={0 | 2}, VGPR1 = rows K={1 | 3}, N = lane&15
    const int arow = lane & 15;
    const int asel = (lane >> 4) << 1;   // 0 for lanes 0-15, 2 for lanes 16-31

    v8f acc_re0 = {};
    v8f acc_im0 = {};
    v8f acc_re1 = {};
    v8f acc_im1 = {};

    for (int chunk = 0; chunk < NCHUNK; ++chunk) {
        const int n0 = chunk * KC;

        __syncthreads();   // protect LDS from previous iteration's readers
        // Stage 32 frames x 80 windowed samples: 2560 floats, 20 per thread.
        #pragma unroll
        for (int i = 0; i < 20; ++i) {
            int e  = i * 128 + tid;        // 0..2559
            int fr = e / KC;               // frame row 0..31
            int nn = e - fr * KC;          // 0..79
            int f  = f_base + fr;
            if (f > NUM_FRAMES - 1) f = NUM_FRAMES - 1;   // clamp: no OOB reads
            ldsA[fr * LDS_STRIDE + nn] =
                sigb[f * FRAME_STEP + n0 + nn] * win[n0 + nn];
        }
        __syncthreads();

        const float* lrow0 = ldsA + arow * LDS_STRIDE + asel;        // 8B aligned
        const float* lrow1 = lrow0 + 16 * LDS_STRIDE;                // frames +16
        // DFT matrices are symmetric: B[n, bin] = dft[bin*800+n] = dft[n*800+bin]
        // -> row-major walk, bins contiguous across lanes (coalesced, L2-hot).
        const float* pc = dcos + (size_t)(n0 + asel) * FRAME_LEN + bin;
        const float* ps = dsin + (size_t)(n0 + asel) * FRAME_LEN + bin;

        #pragma unroll
        for (int kk = 0; kk < KC; kk += 4) {
            v2f a0 = *(const v2f*)(lrow0 + kk);        // ds_load_b64, no conflicts
            v2f a1 = *(const v2f*)(lrow1 + kk);
            v2f bc, bs;
            bc.x = pc[kk * FRAME_LEN];                 // K = kk + asel
            bc.y = pc[(kk + 1) * FRAME_LEN];           // K = kk + asel + 1
            bs.x = ps[kk * FRAME_LEN];
            bs.y = ps[(kk + 1) * FRAME_LEN];
            acc_re0 = __builtin_amdgcn_wmma_f32_16x16x4_f32(
                          false, a0, false, bc, (short)0, acc_re0, false, false);
            acc_im0 = __builtin_amdgcn_wmma_f32_16x16x4_f32(
                          false, a0, false, bs, (short)0, acc_im0, false, false);
            acc_re1 = __builtin_amdgcn_wmma_f32_16x16x4_f32(
                          false, a1, false, bc, (short)0, acc_re1, false, false);
            acc_im1 = __builtin_amdgcn_wmma_f32_16x16x4_f32(
                          false, a1, false, bs, (short)0, acc_im1, false, false);
        }
    }

    // C/D layout: VGPR r -> M = r (lanes 0-15) or r+8 (lanes 16-31), N = lane&15.
    // Output (16,1197,401,2): (re,im) contiguous -> one b64 store per row.
    if (bin < NUM_BINS) {
        const int mhi = (lane >> 4) << 3;              // +8 for upper half-wave
        #pragma unroll
        for (int r = 0; r < 8; ++r) {
            int f0 = f_base + r + mhi;                 // M-subtile 0
            if (f0 < NUM_FRAMES) {
                size_t o = (((size_t)batch * NUM_FRAMES + f0) * NUM_BINS + bin) * 2u;
                float2 v = make_float2(acc_re0[r], acc_im0[r]);
                *(float2*)(out + o) = v;
            }
            int f1 = f0 + 16;                          // M-subtile 1
            if (f1 < NUM_FRAMES) {
                size_t o = (((size_t)batch * NUM_FRAMES + f1) * NUM_BINS + bin) * 2u;
                float2 v = make_float2(acc_re1[r], acc_im1[r]);
                *(float2*)(out + o) = v;
            }
        }
    }
}

extern "C" void kernel_launch(void* const* d_in, const int* in_sizes, int n_in,
                              void* d_out, int out_size, void* d_ws, size_t ws_size,
                              hipStream_t stream) {
    const float* sig  = (const float*)d_in[0];   // (B, 240000) f32
    const float* win  = (const float*)d_in[1];   // (800,)      f32
    const float* dcos = (const float*)d_in[2];   // (800, 800)  f32
    const float* dsin = (const float*)d_in[3];   // (800, 800)  f32
    float* out = (float*)d_out;                  // (B, 1197, 401, 2) f32

    const int batch     = in_sizes[0] / SIG_LEN;            // 16
    const int f_tiles   = (NUM_FRAMES + M_TILE - 1) / M_TILE;  // 38
    const int bin_tiles = (NUM_BINS + 63) / 64;                // 7

    dim3 grid(bin_tiles, f_tiles, batch);
    stft_wmma_f32_kernel<<<grid, 128, 0, stream>>>(sig, win, dcos, dsin, out);
}